// ChunkLayer_63917703299655
// MI455X (gfx1250) — compile-verified
//
#include <hip/hip_runtime.h>
#include <hip/hip_bf16.h>
#include <stdint.h>

// Problem constants (match reference: B=8, L=4096, D=1024, MAX_CHUNKS=128)
#define BB    8
#define LL    4096
#define DD    1024
#define CMAX  128
#define NBUF  4        // LDS ring-buffer depth (rows in flight per block)

// ---------------------------------------------------------------------------
// Kernel 1: per-batch boundary scan. One wave32 per batch.
// Loads all 4096 boundary flags up-front as 32 independent b128 loads per
// lane (one memory-latency round, fully unrolled into registers), then ranks
// boundaries with 4 ballots per 128-token tile.
// Produces pos[b][c] = token index where chunk c starts (c in [0,127]),
// pos[b][128] = end of chunk 127 (p_128 if >=129 boundaries, else L).
// Unreached entries are set to L so empty chunks have start==end (count 0).
// ---------------------------------------------------------------------------
__global__ __launch_bounds__(32) void chunk_scan_kernel(
    const float* __restrict__ bnd, int* __restrict__ pos) {
  const int b    = blockIdx.x;
  const int lane = threadIdx.x;                 // 0..31 (wave32)
  const float4* row4 = reinterpret_cast<const float4*>(bnd + (size_t)b * LL);
  int* p = pos + b * (CMAX + 1);

  // Stage all 16KB of this batch's boundaries into registers: lane holds
  // tokens {k*128 + lane*4 .. +3} for k = 0..31. 32 independent b128 loads.
  float4 vals[32];
#pragma unroll
  for (int k = 0; k < 32; ++k) vals[k] = row4[k * 32 + lane];

  const unsigned lt = (1u << lane) - 1u;
  int base = 0;                                 // #boundaries seen so far
#pragma unroll
  for (int k = 0; k < 32; ++k) {
    const float4 v = vals[k];
    const unsigned b0 = (unsigned)__ballot(v.x > 0.5f);
    const unsigned b1 = (unsigned)__ballot(v.y > 0.5f);
    const unsigned b2 = (unsigned)__ballot(v.z > 0.5f);
    const unsigned b3 = (unsigned)__ballot(v.w > 0.5f);
    // rank prefix: all 4 sub-tokens of every lower lane precede this lane's
    int r = base + __popc(b0 & lt) + __popc(b1 & lt)
                 + __popc(b2 & lt) + __popc(b3 & lt);
    const int t0 = k * 128 + lane * 4;
    if ((b0 >> lane) & 1u) { if (r <= CMAX) p[r] = t0;     ++r; }
    if ((b1 >> lane) & 1u) { if (r <= CMAX) p[r] = t0 + 1; ++r; }
    if ((b2 >> lane) & 1u) { if (r <= CMAX) p[r] = t0 + 2; ++r; }
    if ((b3 >> lane) & 1u) { if (r <= CMAX) p[r] = t0 + 3; ++r; }
    base += __popc(b0) + __popc(b1) + __popc(b2) + __popc(b3);
  }
  // Fill the tail (chunks that never start) with L.
  for (int c = base + lane; c <= CMAX; c += 32) p[c] = LL;
}

// ---------------------------------------------------------------------------
// Kernel 2: one block per (batch, chunk). 256 threads; thread t owns the
// 16-byte slice [t*4, t*4+4) of the D=1024 row. Rows stream HBM -> LDS via
// async b128 copies in a 4-deep ring; each wave's async instruction writes
// exactly the LDS bytes its own lanes later read, so per-wave
// s_wait_asynccnt is the only synchronization needed (async loads complete
// in order within a wave). Steady-state loop uses a fixed wait immediate;
// the <=NBUF-row tail drains with asynccnt==0.
// ---------------------------------------------------------------------------
__global__ __launch_bounds__(256) void chunk_mean_kernel(
    const float* __restrict__ x, const int* __restrict__ pos,
    float* __restrict__ out) {
  const int bc = blockIdx.x;
  const int b  = bc >> 7;                       // /128
  const int c  = bc & (CMAX - 1);               // %128
  const int t  = threadIdx.x;

  const int s = pos[b * (CMAX + 1) + c];
  const int e = pos[b * (CMAX + 1) + c + 1];
  const int n = e - s;                          // tokens in this chunk

  __shared__ __align__(16) float lbuf[NBUF * DD];
  // gfx12 generic->LDS: low 32 bits of a generic LDS pointer are the
  // wave-relative LDS byte address the async instruction wants in VDST.
  const unsigned lds_t = (unsigned)(uintptr_t)(&lbuf[0]) + (unsigned)(t * 16);
  const float*   gbase = x + ((size_t)b * LL + s) * DD + t * 4;

  float a0 = 0.f, a1 = 0.f, a2 = 0.f, a3 = 0.f;

  // Prologue: fill the pipeline with up to NBUF rows.
  const int npre = n < NBUF ? n : NBUF;
  for (int k = 0; k < npre; ++k) {
    const float*   src = gbase + (size_t)k * DD;
    const unsigned dst = lds_t + (unsigned)(k * (DD * 4));
    asm volatile("global_load_async_to_lds_b128 %0, %1, off"
                 :: "v"(dst), "v"(src) : "memory");
  }

  int i = 0;
  // Steady state: rows i..i+NBUF-1 in flight; in-order completion means
  // asynccnt<=NBUF-1 guarantees row i has landed in LDS.
  for (; i < n - NBUF; ++i) {
    asm volatile("s_wait_asynccnt %0" :: "n"(NBUF - 1) : "memory");
    const int slot = i & (NBUF - 1);
    const float4 v =
        *reinterpret_cast<const float4*>(&lbuf[slot * DD + t * 4]); // ds_load_b128
    a0 += v.x; a1 += v.y; a2 += v.z; a3 += v.w;
    // Refill the slot we just consumed (issue is in-order; the adds above
    // force the dscnt wait before this async op can overwrite the slot).
    const float*   src = gbase + (size_t)(i + NBUF) * DD;
    const unsigned dst = lds_t + (unsigned)(slot * (DD * 4));
    asm volatile("global_load_async_to_lds_b128 %0, %1, off"
                 :: "v"(dst), "v"(src) : "memory");
  }
  // Drain tail (also the whole loop when n <= NBUF).
  for (; i < n; ++i) {
    asm volatile("s_wait_asynccnt 0" ::: "memory");
    const int slot = i & (NBUF - 1);
    const float4 v =
        *reinterpret_cast<const float4*>(&lbuf[slot * DD + t * 4]);
    a0 += v.x; a1 += v.y; a2 += v.z; a3 += v.w;
  }

  const float inv = (n > 0) ? (1.0f / (float)n) : 0.0f;
  float4 r;
  r.x = a0 * inv; r.y = a1 * inv; r.z = a2 * inv; r.w = a3 * inv;
  *reinterpret_cast<float4*>(&out[(size_t)bc * DD + t * 4]) = r;

  // Counts appended after the means (float storage; counts <= 4096 are exact).
  if (t == 0) out[(size_t)BB * CMAX * DD + bc] = (float)n;
}

// ---------------------------------------------------------------------------
// Launch: inputs per reference order: d_in[0]=x [8,4096,1024] f32,
// d_in[1]=boundaries [8,4096] f32. d_out = means [8,128,1024] ++ cnts [8,128].
// d_ws: int pos[8][129] (4128 bytes).
// ---------------------------------------------------------------------------
extern "C" void kernel_launch(void* const* d_in, const int* in_sizes, int n_in,
                              void* d_out, int out_size, void* d_ws, size_t ws_size,
                              hipStream_t stream) {
  (void)in_sizes; (void)n_in; (void)out_size; (void)ws_size;
  const float* x   = (const float*)d_in[0];
  const float* bnd = (const float*)d_in[1];
  float* out = (float*)d_out;
  int*   pos = (int*)d_ws;

  chunk_scan_kernel<<<BB, 32, 0, stream>>>(bnd, pos);
  chunk_mean_kernel<<<BB * CMAX, 256, 0, stream>>>(x, pos, out);
}